// GATBlock_58566174048908
// MI455X (gfx1250) — compile-verified
//
#include <hip/hip_runtime.h>

typedef __attribute__((ext_vector_type(2))) float v2f;
typedef __attribute__((ext_vector_type(8))) float v8f;

#define IN_C      256
#define OUT_C     128
#define HEADS     2
#define HC        (HEADS * OUT_C)   // 256 columns in both layer GEMMs
#define NEG_SLOPE 0.2f

// ---------- helpers ----------
__device__ __forceinline__ float lrelu(float a) { return a > 0.f ? a : NEG_SLOPE * a; }

// monotonic unsigned encoding of float for atomicMax-based segment max
__device__ __forceinline__ unsigned encf(float f) {
    unsigned u = __float_as_uint(f);
    return (u & 0x80000000u) ? ~u : (u | 0x80000000u);
}
__device__ __forceinline__ float decf(unsigned u) {
    unsigned r = (u & 0x80000000u) ? (u & 0x7FFFFFFFu) : ~u;
    return __uint_as_float(r);
}
#define ENC_NEG_INF 0x007FFFFFu   // encf(-inf)

// ---------- fills ----------
__global__ void fill_f32(float* p, float v, long long n) {
    long long i = (long long)blockIdx.x * blockDim.x + threadIdx.x;
    if (i < n) p[i] = v;
}
__global__ void fill_u32(unsigned* p, unsigned v, long long n) {
    long long i = (long long)blockIdx.x * blockDim.x + threadIdx.x;
    if (i < n) p[i] = v;
}

// ---------- fp32 WMMA GEMM: C[M x 256] = A[M x K] (lda) @ B[K x 256] ----------
// One block (8 waves / 256 threads) covers a full 16-row x 256-col strip, so the
// A strip is read from HBM exactly once. Each wave owns two 16x16 column tiles
// (cols wave*16 and wave*16+128) and shares one A operand across both WMMAs.
// A strip is staged into LDS via gfx1250 async copy (GLOBAL_LOAD_ASYNC_TO_LDS_B128,
// ASYNCcnt-tracked, bypasses VGPRs). B (<=256KB) stays L2-hot.
__global__ void __launch_bounds__(256)
gemm_wmma_f32(const float* __restrict__ A, int lda, int K,
              const float* __restrict__ B,
              float* __restrict__ C, int M)
{
    __shared__ float ldsA[16 * 260];            // K<=256; stride K+4 keeps 16B align + bank spread
    const int lane    = threadIdx.x & 31;
    const int wave    = threadIdx.x >> 5;
    const int row0    = blockIdx.x << 4;
    const int lstride = K + 4;
    const int col0    = wave << 4;              // first tile: cols [col0, col0+16)
    const int col1    = col0 + 128;             // second tile

    // ---- async-stage the 16 x K A strip into LDS (16B per lane per issue) ----
    {
        const int kv = K >> 2;                  // float4s per row
        for (int idx = threadIdx.x; idx < 16 * kv; idx += 256) {
            int r  = idx / kv;
            int k4 = idx - r * kv;
            int gr = row0 + r;
            if (gr >= M) gr = M - 1;            // rows are independent in GEMM: clamp is safe
            unsigned lds_off = (unsigned)(size_t)(ldsA + r * lstride + (k4 << 2));
            const float* gaddr = A + (size_t)gr * lda + (k4 << 2);
            asm volatile("global_load_async_to_lds_b128 %0, %1, off"
                         :: "v"(lds_off), "v"(gaddr) : "memory");
        }
        asm volatile("s_wait_asynccnt 0x0" ::: "memory");
        __syncthreads();
    }

    v8f acc0 = {}, acc1 = {};
    const int n15   = lane & 15;          // M index for A, N index for B/C
    const int khalf = (lane >> 4) << 1;   // lanes 0-15 -> K+{0,1}; lanes 16-31 -> K+{2,3}

    #pragma unroll 4
    for (int k0 = 0; k0 < K; k0 += 4) {
        v2f a, b0, b1;
        // A 16x4 f32 layout (ISA 7.12.2): lane half selects K pair, M = lane&15
        a.x = ldsA[n15 * lstride + k0 + khalf + 0];
        a.y = ldsA[n15 * lstride + k0 + khalf + 1];
        // B 4x16 f32 layout: row K striped across lanes within a VGPR
        const float* Bk = B + (size_t)(k0 + khalf) * HC + n15;
        b0.x = Bk[col0];
        b0.y = Bk[HC + col0];
        b1.x = Bk[col1];
        b1.y = Bk[HC + col1];
        if (k0 + 8 < K)
            __builtin_prefetch(Bk + 8 * HC + col0, 0, 1);
        acc0 = __builtin_amdgcn_wmma_f32_16x16x4_f32(
                   false, a, false, b0, (short)0, acc0, false, false);
        acc1 = __builtin_amdgcn_wmma_f32_16x16x4_f32(
                   false, a, false, b1, (short)0, acc1, false, false);
    }

    // C 16x16 f32 layout: vgpr r -> M = r + 8*(lane>=16), N = lane&15
    const int rbase = row0 + ((lane >> 4) << 3);
    #pragma unroll
    for (int r = 0; r < 8; ++r) {
        int row = rbase + r;
        if (row < M) {
            C[(size_t)row * HC + col0 + n15] = acc0[r];
            C[(size_t)row * HC + col1 + n15] = acc1[r];
        }
    }
}

// ---------- per-(node, head) attention scores: wave32 dot products ----------
__global__ void __launch_bounds__(256)
attn_scores(const float* __restrict__ xs,
            const float* __restrict__ att_src, const float* __restrict__ att_dst,
            float* __restrict__ sA, float* __restrict__ sD, int N)
{
    int gw   = (blockIdx.x * blockDim.x + threadIdx.x) >> 5;
    int lane = threadIdx.x & 31;
    int n = gw >> 1, h = gw & 1;
    if (n >= N) return;

    const float* xp = xs + (size_t)n * HC + h * OUT_C;
    const float* as = att_src + h * OUT_C;
    const float* ad = att_dst + h * OUT_C;
    float s = 0.f, d = 0.f;
    #pragma unroll
    for (int c = lane; c < OUT_C; c += 32) {
        float v = xp[c];
        s += v * as[c];
        d += v * ad[c];
    }
    #pragma unroll
    for (int off = 16; off > 0; off >>= 1) {
        s += __shfl_down(s, off, 32);
        d += __shfl_down(d, off, 32);
    }
    if (lane == 0) { sA[n * 2 + h] = s; sD[n * 2 + h] = d; }
}

// ---------- segment max over dst (monotonic-uint atomicMax) ----------
__global__ void edge_max(const int* __restrict__ ei, int E, int Ep,
                         const float* __restrict__ sA, const float* __restrict__ sD,
                         unsigned* __restrict__ enc)
{
    int e = blockIdx.x * blockDim.x + threadIdx.x;
    if (e >= Ep) return;
    int src = (e < E) ? ei[e]     : (e - E);   // self-loops appended
    int dst = (e < E) ? ei[E + e] : (e - E);
    #pragma unroll
    for (int h = 0; h < HEADS; ++h) {
        float alpha = lrelu(sA[src * 2 + h] + sD[dst * 2 + h]);
        atomicMax(&enc[dst * 2 + h], encf(alpha));
    }
}

__global__ void decode_max(const unsigned* __restrict__ enc, float* __restrict__ amax, int n)
{
    int i = blockIdx.x * blockDim.x + threadIdx.x;
    if (i >= n) return;
    float v = decf(enc[i]);
    amax[i] = (v > -3.0e38f && v < 3.0e38f) ? v : 0.f;  // where(isfinite, amax, 0)
}

// ---------- softmax denominator ----------
__global__ void edge_expsum(const int* __restrict__ ei, int E, int Ep,
                            const float* __restrict__ sA, const float* __restrict__ sD,
                            const float* __restrict__ amax, float* __restrict__ denom)
{
    int e = blockIdx.x * blockDim.x + threadIdx.x;
    if (e >= Ep) return;
    int src = (e < E) ? ei[e]     : (e - E);
    int dst = (e < E) ? ei[E + e] : (e - E);
    #pragma unroll
    for (int h = 0; h < HEADS; ++h) {
        float alpha = lrelu(sA[src * 2 + h] + sD[dst * 2 + h]);
        atomicAdd(&denom[dst * 2 + h], expf(alpha - amax[dst * 2 + h]));
    }
}

// ---------- weighted message scatter: one wave per edge, L2-resident f32 atomics ----------
__global__ void __launch_bounds__(256)
edge_aggregate(const int* __restrict__ ei, int E, int Ep,
               const float* __restrict__ sA, const float* __restrict__ sD,
               const float* __restrict__ amax, const float* __restrict__ denom,
               const float* __restrict__ xs, float* __restrict__ accum)
{
    int gw   = (blockIdx.x * blockDim.x + threadIdx.x) >> 5;
    int lane = threadIdx.x & 31;
    if (gw >= Ep) return;
    int e   = gw;
    int src = (e < E) ? ei[e]     : (e - E);
    int dst = (e < E) ? ei[E + e] : (e - E);

    float coef[HEADS];
    #pragma unroll
    for (int h = 0; h < HEADS; ++h) {
        float alpha = lrelu(sA[src * 2 + h] + sD[dst * 2 + h]);
        float ex    = expf(alpha - amax[dst * 2 + h]);
        coef[h]     = ex / (denom[dst * 2 + h] + 1e-16f);
    }
    const float* xsrc = xs    + (size_t)src * HC;
    float*       adst = accum + (size_t)dst * HC;
    #pragma unroll
    for (int c = lane; c < OUT_C; c += 32) {
        atomicAdd(&adst[c],         xsrc[c]         * coef[0]);
        atomicAdd(&adst[OUT_C + c], xsrc[OUT_C + c] * coef[1]);
    }
}

// ---------- head mean + bias ----------
__global__ void head_mean_inplace(float* __restrict__ acc, const float* __restrict__ bias, int N)
{
    long long i = (long long)blockIdx.x * blockDim.x + threadIdx.x;
    if (i >= (long long)N * OUT_C) return;
    int n = (int)(i >> 7), c = (int)(i & 127);
    float v = 0.5f * (acc[(size_t)n * HC + c] + acc[(size_t)n * HC + OUT_C + c]) + bias[c];
    acc[(size_t)n * HC + c] = v;   // result overlays cols 0..127 -> next GEMM uses lda=HC
}
__global__ void head_mean_out(const float* __restrict__ acc, const float* __restrict__ bias,
                              float* __restrict__ out, int N)
{
    long long i = (long long)blockIdx.x * blockDim.x + threadIdx.x;
    if (i >= (long long)N * OUT_C) return;
    int n = (int)(i >> 7), c = (int)(i & 127);
    out[i] = 0.5f * (acc[(size_t)n * HC + c] + acc[(size_t)n * HC + OUT_C + c]) + bias[c];
}

// ---------- host launch ----------
extern "C" void kernel_launch(void* const* d_in, const int* in_sizes, int n_in,
                              void* d_out, int out_size, void* d_ws, size_t ws_size,
                              hipStream_t stream)
{
    (void)n_in; (void)out_size; (void)ws_size;
    const float* x   = (const float*)d_in[0];
    const int*   ei  = (const int*)  d_in[1];
    const float* W0  = (const float*)d_in[2];
    const float* as0 = (const float*)d_in[3];
    const float* ad0 = (const float*)d_in[4];
    const float* b0  = (const float*)d_in[5];
    const float* W1  = (const float*)d_in[6];
    const float* as1 = (const float*)d_in[7];
    const float* ad1 = (const float*)d_in[8];
    const float* b1  = (const float*)d_in[9];
    float* out = (float*)d_out;

    const int N  = in_sizes[0] / IN_C;
    const int E  = in_sizes[1] / 2;
    const int Ep = E + N;

    char* ws = (char*)d_ws;
    auto alloc = [&](size_t bytes) -> char* {
        char* p = ws; ws += (bytes + 255) & ~(size_t)255; return p;
    };
    float*    xs    = (float*)   alloc((size_t)N * HC * 4);
    float*    accum = (float*)   alloc((size_t)N * HC * 4);
    float*    sA    = (float*)   alloc((size_t)N * 2 * 4);
    float*    sD    = (float*)   alloc((size_t)N * 2 * 4);
    unsigned* enc   = (unsigned*)alloc((size_t)N * 2 * 4);
    float*    amax  = (float*)   alloc((size_t)N * 2 * 4);
    float*    denom = (float*)   alloc((size_t)N * 2 * 4);

    const long long nacc = (long long)N * HC;
    const long long n2   = (long long)N * 2;

    auto run_layer = [&](const float* A, int lda, int K, const float* W,
                         const float* att_s, const float* att_d) {
        gemm_wmma_f32<<<(N + 15) / 16, 256, 0, stream>>>(A, lda, K, W, xs, N);
        // zero/init after GEMM so layer-1 can reuse accum (its GEMM input) as accumulator
        fill_f32<<<(unsigned)((nacc + 255) / 256), 256, 0, stream>>>(accum, 0.f, nacc);
        fill_f32<<<(unsigned)((n2 + 255) / 256), 256, 0, stream>>>(denom, 0.f, n2);
        fill_u32<<<(unsigned)((n2 + 255) / 256), 256, 0, stream>>>(enc, ENC_NEG_INF, n2);

        long long scoreThreads = (long long)N * 2 * 32;
        attn_scores<<<(unsigned)((scoreThreads + 255) / 256), 256, 0, stream>>>(
            xs, att_s, att_d, sA, sD, N);
        edge_max<<<(unsigned)((Ep + 255) / 256), 256, 0, stream>>>(ei, E, Ep, sA, sD, enc);
        decode_max<<<(unsigned)((n2 + 255) / 256), 256, 0, stream>>>(enc, amax, (int)n2);
        edge_expsum<<<(unsigned)((Ep + 255) / 256), 256, 0, stream>>>(
            ei, E, Ep, sA, sD, amax, denom);
        long long aggThreads = (long long)Ep * 32;
        edge_aggregate<<<(unsigned)((aggThreads + 255) / 256), 256, 0, stream>>>(
            ei, E, Ep, sA, sD, amax, denom, xs, accum);
    };

    // layer 0: x[N,256] @ W0[256,256]
    run_layer(x, IN_C, IN_C, W0, as0, ad0);
    head_mean_inplace<<<(unsigned)(((long long)N * OUT_C + 255) / 256), 256, 0, stream>>>(
        accum, b0, N);
    // layer 1: h[N,128] (stride HC inside accum) @ W1[128,256]
    run_layer(accum, HC, OUT_C, W1, as1, ad1);
    head_mean_out<<<(unsigned)(((long long)N * OUT_C + 255) / 256), 256, 0, stream>>>(
        accum, b1, out, N);
}